// GCNReg_15152644620458
// MI455X (gfx1250) — compile-verified
//
#include <hip/hip_runtime.h>
#include <hip/hip_bf16.h>
#include <math.h>

#define NN 100000      // nodes
#define NE 1200000     // edges
#define HID 64
#define NG 512         // graphs
#define LN_EPS 1e-5f
#define NORM_EPS 1e-12f

typedef __attribute__((ext_vector_type(2))) float v2f;
typedef __attribute__((ext_vector_type(8))) float v8f;

#if defined(__AMDGCN__) && __has_builtin(__builtin_amdgcn_wmma_f32_16x16x4_f32)
#define USE_WMMA 1
#else
#define USE_WMMA 0
#endif

// ---------------------------------------------------------------- utilities
__global__ void zero_f32(float* __restrict__ p, long n) {
  long i = (long)blockIdx.x * blockDim.x + threadIdx.x;
  long stride = (long)gridDim.x * blockDim.x;
  for (; i < n; i += stride) p[i] = 0.0f;
}

// ---------------------------------------------------------------- degrees
__global__ void degree_kernel(const int* __restrict__ src, const int* __restrict__ dst,
                              float* __restrict__ deg_out, float* __restrict__ deg_in, int E) {
  int e = blockIdx.x * blockDim.x + threadIdx.x;
  if (e < E) {
    atomicAdd(&deg_out[src[e]], 1.0f);
    atomicAdd(&deg_in[dst[e]], 1.0f);
  }
}

__global__ void invsqrt_clamp_kernel(float* __restrict__ p, int n) {
  int i = blockIdx.x * blockDim.x + threadIdx.x;
  if (i < n) p[i] = rsqrtf(fmaxf(p[i], 1.0f));
}

// --------------------------------------------------- edge gather/scatter-add
// one wave (32 lanes) per edge; each lane moves 2 consecutive f32 features.
// msg = (x * ns)[src] scattered into agg[dst] via f32 atomics (L2-resident).
__global__ void edge_scatter_kernel(const float* __restrict__ x, const float* __restrict__ ns,
                                    const int* __restrict__ src, const int* __restrict__ dst,
                                    float* __restrict__ agg, int E) {
  long g = (long)blockIdx.x * blockDim.x + threadIdx.x;
  long e = g >> 5;
  int lane = (int)(g & 31);
  if (e >= E) return;
  int s = src[e];
  int d = dst[e];
  float sc = ns[s];
  const float* xp = x + (long)s * HID + lane * 2;
  float vx = xp[0] * sc;
  float vy = xp[1] * sc;
  float* ap = agg + (long)d * HID + lane * 2;
  atomicAdd(ap + 0, vx);
  atomicAdd(ap + 1, vy);
}

// --------------------------- WMMA GEMM (fp32 16x16x4) + bias + LayerNorm + ReLU
// One wave per block, 16 output rows x 64 output cols.
// out[m, :] = relu( LN( (A[m,:] * rowscale[m]) @ W + bias ) * gamma + beta )
template <int K>
__global__ __launch_bounds__(32) void wmma_gemm_ln_relu(
    const float* __restrict__ A, const float* __restrict__ rowscale,
    const float* __restrict__ W, const float* __restrict__ bias,
    const float* __restrict__ gamma, const float* __restrict__ beta,
    float* __restrict__ out) {
  const int lane = threadIdx.x;        // 0..31
  const int half = lane >> 4;          // 0 or 1
  const int lr   = lane & 15;          // 0..15
  const int m0   = blockIdx.x * 16;
  const int m    = m0 + lr;            // A row for this lane's A-fragment
  const int koff = half * 2;           // lanes<16: K+0,K+1 ; lanes>=16: K+2,K+3
  const float s  = rowscale ? rowscale[m] : 1.0f;

  v8f c0 = {}, c1 = {}, c2 = {}, c3 = {};

  for (int k = 0; k < K; k += 4) {
#if USE_WMMA
    v2f a, b;
    {
      const float* ap = A + (long)m * K + k + koff;
      a[0] = ap[0] * s;
      a[1] = ap[1] * s;
    }
    const float* w0 = W + (long)(k + koff) * HID;      // row k+koff   of W
    const float* w1 = w0 + HID;                        // row k+koff+1 of W
    b[0] = w0[lr];      b[1] = w1[lr];
    c0 = __builtin_amdgcn_wmma_f32_16x16x4_f32(false, a, false, b, (short)0, c0, false, false);
    b[0] = w0[16 + lr]; b[1] = w1[16 + lr];
    c1 = __builtin_amdgcn_wmma_f32_16x16x4_f32(false, a, false, b, (short)0, c1, false, false);
    b[0] = w0[32 + lr]; b[1] = w1[32 + lr];
    c2 = __builtin_amdgcn_wmma_f32_16x16x4_f32(false, a, false, b, (short)0, c2, false, false);
    b[0] = w0[48 + lr]; b[1] = w1[48 + lr];
    c3 = __builtin_amdgcn_wmma_f32_16x16x4_f32(false, a, false, b, (short)0, c3, false, false);
#else
    // scalar fallback producing the identical C lane layout (host pass / no-builtin)
    for (int r = 0; r < 8; ++r) {
      int row = m0 + r + half * 8;
      float sr = rowscale ? rowscale[row] : 1.0f;
      for (int kk = 0; kk < 4; ++kk) {
        float av = A[(long)row * K + k + kk] * sr;
        const float* wr = W + (long)(k + kk) * HID;
        c0[r] += av * wr[lr];
        c1[r] += av * wr[16 + lr];
        c2[r] += av * wr[32 + lr];
        c3[r] += av * wr[48 + lr];
      }
    }
#endif
  }

  // C layout: VGPR r, lane l -> row = m0 + r + 8*(l/16), col = 16*tile + (l%16)
  __shared__ float tile[16][HID + 4];
  const float b0 = bias[lr], b1 = bias[16 + lr], b2 = bias[32 + lr], b3 = bias[48 + lr];
#pragma unroll
  for (int r = 0; r < 8; ++r) {
    int row = r + half * 8;
    tile[row][lr]      = c0[r] + b0;
    tile[row][16 + lr] = c1[r] + b1;
    tile[row][32 + lr] = c2[r] + b2;
    tile[row][48 + lr] = c3[r] + b3;
  }
  __syncthreads();

  // LayerNorm over 64 features: 2 lanes per row (lane lr and lane lr+16).
  const int row   = lr;
  const int fbase = half * 32;
  float sum = 0.0f, sq = 0.0f;
#pragma unroll
  for (int i = 0; i < 32; ++i) {
    float v = tile[row][fbase + i];
    sum += v;
    sq  += v * v;
  }
  sum += __shfl_xor(sum, 16, 32);
  sq  += __shfl_xor(sq, 16, 32);
  const float mu   = sum * (1.0f / 64.0f);
  const float var  = sq * (1.0f / 64.0f) - mu * mu;
  const float rstd = rsqrtf(var + LN_EPS);

  float* orow = out + (long)(m0 + row) * HID;
#pragma unroll
  for (int i = 0; i < 32; ++i) {
    int f = fbase + i;
    float v = (tile[row][f] - mu) * rstd * gamma[f] + beta[f];
    orow[f] = fmaxf(v, 0.0f);
  }
}

// ---------------------------------------------------------------- pooling
// one wave per node; sum + count + max (post-ReLU values >= 0 -> uint-bit max OK)
__global__ void pool_kernel(const float* __restrict__ h2, const int* __restrict__ gids,
                            float* __restrict__ ssum, unsigned* __restrict__ maxb,
                            float* __restrict__ cnt, int N) {
  long g = (long)blockIdx.x * blockDim.x + threadIdx.x;
  long n = g >> 5;
  int lane = (int)(g & 31);
  if (n >= N) return;
  int gid = gids[n];
  const float* hp = h2 + n * HID + lane * 2;
  float vx = hp[0], vy = hp[1];
  atomicAdd(&ssum[(long)gid * HID + lane * 2 + 0], vx);
  atomicAdd(&ssum[(long)gid * HID + lane * 2 + 1], vy);
  atomicMax(&maxb[(long)gid * HID + lane * 2 + 0], __float_as_uint(vx));
  atomicMax(&maxb[(long)gid * HID + lane * 2 + 1], __float_as_uint(vy));
  if (lane == 0) atomicAdd(&cnt[gid], 1.0f);
}

// build X = concat(l2norm(mean), l2norm(max)) -> [NG, 128]
__global__ void build_x_kernel(const float* __restrict__ ssum, const unsigned* __restrict__ maxb,
                               const float* __restrict__ cnt, float* __restrict__ X) {
  int g = blockIdx.x * blockDim.x + threadIdx.x;
  if (g >= NG) return;
  float inv_c = 1.0f / fmaxf(cnt[g], 1.0f);
  float sm = 0.0f, sx = 0.0f;
  for (int f = 0; f < HID; ++f) {
    float mv = ssum[(long)g * HID + f] * inv_c;
    sm += mv * mv;
    float xv = __uint_as_float(maxb[(long)g * HID + f]);
    sx += xv * xv;
  }
  float im = 1.0f / fmaxf(sqrtf(sm), NORM_EPS);
  float ix = 1.0f / fmaxf(sqrtf(sx), NORM_EPS);
  for (int f = 0; f < HID; ++f) {
    X[(long)g * (2 * HID) + f]       = ssum[(long)g * HID + f] * inv_c * im;
    X[(long)g * (2 * HID) + HID + f] = __uint_as_float(maxb[(long)g * HID + f]) * ix;
  }
}

// out = t2 @ Wc3 + bc3   (Wc3 is [64,1])
__global__ void final_linear_kernel(const float* __restrict__ t2, const float* __restrict__ Wc3,
                                    const float* __restrict__ bc3, float* __restrict__ out) {
  int g = blockIdx.x * blockDim.x + threadIdx.x;
  if (g >= NG) return;
  float acc = bc3[0];
  for (int i = 0; i < HID; ++i) acc += t2[(long)g * HID + i] * Wc3[i];
  out[g] = acc;
}

// ---------------------------------------------------------------- launcher
extern "C" void kernel_launch(void* const* d_in, const int* in_sizes, int n_in,
                              void* d_out, int out_size, void* d_ws, size_t ws_size,
                              hipStream_t stream) {
  const float* h   = (const float*)d_in[0];
  const int* src   = (const int*)d_in[1];
  const int* dst   = (const int*)d_in[2];
  const int* gids  = (const int*)d_in[3];
  const float* W1  = (const float*)d_in[4];
  const float* b1  = (const float*)d_in[5];
  const float* W2  = (const float*)d_in[6];
  const float* b2  = (const float*)d_in[7];
  const float* g1  = (const float*)d_in[8];
  const float* be1 = (const float*)d_in[9];
  const float* g2  = (const float*)d_in[10];
  const float* be2 = (const float*)d_in[11];
  const float* g3  = (const float*)d_in[12];
  const float* be3 = (const float*)d_in[13];
  const float* g4  = (const float*)d_in[14];
  const float* be4 = (const float*)d_in[15];
  const float* Wc1 = (const float*)d_in[16];
  const float* bc1 = (const float*)d_in[17];
  const float* Wc2 = (const float*)d_in[18];
  const float* bc2 = (const float*)d_in[19];
  const float* Wc3 = (const float*)d_in[20];
  const float* bc3 = (const float*)d_in[21];
  float* out = (float*)d_out;

  // workspace carve-up (floats)
  float* ws = (float*)d_ws;
  float* ns   = ws;                     // [NN]  deg_out -> deg_out^-1/2
  float* nd   = ns + NN;                // [NN]  deg_in  -> deg_in^-1/2
  float* agg  = nd + NN;                // [NN*64]
  float* h1   = agg + (long)NN * HID;   // [NN*64]  (reused as h2)
  float* ssum = h1 + (long)NN * HID;    // [NG*64]
  unsigned* maxb = (unsigned*)(ssum + NG * HID); // [NG*64]
  float* cnt  = (float*)(maxb + NG * HID);       // [NG]
  float* X    = cnt + NG;               // [NG*128]
  float* t1   = X + NG * 2 * HID;       // [NG*64]
  float* t2   = t1 + NG * HID;          // [NG*64]

  const int E = in_sizes[1];
  const int N = NN;

  // 1) degrees -> symmetric norms
  zero_f32<<<2048, 256, 0, stream>>>(ns, 2L * N);
  degree_kernel<<<(E + 255) / 256, 256, 0, stream>>>(src, dst, ns, nd, E);
  invsqrt_clamp_kernel<<<(2 * N + 255) / 256, 256, 0, stream>>>(ns, 2 * N);

  // 2) conv1: scatter then WMMA GEMM + LN + ReLU
  zero_f32<<<4096, 256, 0, stream>>>(agg, (long)N * HID);
  edge_scatter_kernel<<<(int)(((long)E * 32 + 255) / 256), 256, 0, stream>>>(h, ns, src, dst, agg, E);
  wmma_gemm_ln_relu<HID><<<N / 16, 32, 0, stream>>>(agg, nd, W1, b1, g1, be1, h1);

  // 3) conv2
  zero_f32<<<4096, 256, 0, stream>>>(agg, (long)N * HID);
  edge_scatter_kernel<<<(int)(((long)E * 32 + 255) / 256), 256, 0, stream>>>(h1, ns, src, dst, agg, E);
  wmma_gemm_ln_relu<HID><<<N / 16, 32, 0, stream>>>(agg, nd, W2, b2, g2, be2, h1);

  // 4) readout: mean + max pooling, l2norm, concat
  zero_f32<<<512, 256, 0, stream>>>(ssum, (long)NG * HID * 2 + NG); // ssum, maxb(0 bits=0.0f), cnt
  pool_kernel<<<(int)(((long)N * 32 + 255) / 256), 256, 0, stream>>>(h1, gids, ssum, maxb, cnt, N);
  build_x_kernel<<<(NG + 255) / 256, 256, 0, stream>>>(ssum, maxb, cnt, X);

  // 5) classifier head
  wmma_gemm_ln_relu<2 * HID><<<NG / 16, 32, 0, stream>>>(X, nullptr, Wc1, bc1, g3, be3, t1);
  wmma_gemm_ln_relu<HID><<<NG / 16, 32, 0, stream>>>(t1, nullptr, Wc2, bc2, g4, be4, t2);
  final_linear_kernel<<<(NG + 255) / 256, 256, 0, stream>>>(t2, Wc3, bc3, out);
}